// SPL_ComputeWithTrace_4131758539314
// MI455X (gfx1250) — compile-verified
//
#include <hip/hip_runtime.h>
#include <hip/hip_bf16.h>
#include <math.h>

typedef __attribute__((ext_vector_type(2))) float v2f;
typedef __attribute__((ext_vector_type(8))) float v8f;

#define NSLICE 48                    // B*C
#define HDIM 512
#define WDIM 512
#define BANDROWS 16
#define NBANDS (HDIM / BANDROWS)     // 32
#define BANDS_PER_BLOCK 2
#define CHUNKS (NBANDS / BANDS_PER_BLOCK)  // 16 blocks per slice -> 768 blocks total
#define LSTR 516                     // padded LDS row stride: 516%64==4 -> conflict-free, 516*4%16==0 -> b128-aligned
#define EPSF 1e-12f
#define PLANE_SMALL (NSLICE * WDIM)            // 24576 floats per stat plane (atomic path)
#define PLANE_BIG   (NSLICE * CHUNKS * WDIM)   // 393216 floats per stat plane (store path)

__device__ __forceinline__ float vext8(v8f v, int i) {
  float r = v[0];
#pragma unroll
  for (int k = 1; k < 8; ++k) { if (i == k) r = v[k]; }
  return r;
}

__device__ __forceinline__ void gatomAdd(float* p, float v) {
  unsafeAtomicAdd(p, v);   // global_atomic_add_f32
}

__global__ void __launch_bounds__(256) init_ws_kernel(float* ws, int n) {
  int i = blockIdx.x * 256 + threadIdx.x;
  if (i < n) ws[i] = 0.f;
}

// One block = BANDS_PER_BLOCK consecutive 16-row bands of one (b,c) slice.
// Single fused pass over HBM: per-row stats via fp32 WMMA Gram diagonals
// (X*R^T, X*X^T, R*R^T), per-column partial stats via VALU, both from one
// LDS staging per band; next band's loads are prefetched into registers.
// ATOMIC_COLS=true : accumulate column partials with global f32 atomics.
// ATOMIC_COLS=false: store per-(slice,chunk) partial planes (no atomics).
template <bool ATOMIC_COLS>
__global__ void __launch_bounds__(256) band_kernel(
    const float* __restrict__ x, const float* __restrict__ ref,
    float* __restrict__ colOut, float* __restrict__ rowSum) {
  extern __shared__ float smem[];
  float* xs  = smem;                     // 16*516
  float* rs  = smem + BANDROWS * LSTR;   // 16*516
  float* acc = rs + BANDROWS * LSTR;     // 48: [0..15]=dot [16..31]=xx [32..47]=rr

  const int tid   = threadIdx.x;
  const int slice = blockIdx.x / CHUNKS;
  const int chunk = blockIdx.x % CHUNKS;
  const int band0 = chunk * BANDS_PER_BLOCK;
  const size_t sliceBase = (size_t)slice * HDIM * WDIM;

  if (tid < 48) acc[tid] = 0.f;

  // ---- prefetch first band into registers (16 b128 loads in flight) ----
  float4 px[8], pr[8];
  {
    const float4* xg = (const float4*)(x + sliceBase + (size_t)band0 * BANDROWS * WDIM);
    const float4* rg = (const float4*)(ref + sliceBase + (size_t)band0 * BANDROWS * WDIM);
#pragma unroll
    for (int c = 0; c < 8; ++c) { px[c] = xg[tid + c * 256]; pr[c] = rg[tid + c * 256]; }
  }

  const int wave = tid >> 5;
  const int lane = tid & 31;
  const int arow = lane & 15;            // M (=N) row this lane feeds
  const int koff = (lane >> 4) << 1;     // lanes 16-31 hold K=2,3 of each chunk
  const int cbase = wave * 64;           // wave's K (column) range

  float csd[2] = {0.f, 0.f}, csx[2] = {0.f, 0.f}, csr[2] = {0.f, 0.f};
  float rowTermAcc = 0.f;                // live in threads 0..15

  for (int b = 0; b < BANDS_PER_BLOCK; ++b) {
    // ---- stage prefetched band into LDS ----
#pragma unroll
    for (int c = 0; c < 8; ++c) {
      const int i = tid + c * 256;
      const int row = i >> 7;            // 128 float4 per row
      const int col = (i & 127) << 2;
      *(float4*)&xs[row * LSTR + col] = px[c];
      *(float4*)&rs[row * LSTR + col] = pr[c];
    }
    __syncthreads();

    // ---- prefetch next band while computing this one ----
    if (b + 1 < BANDS_PER_BLOCK) {
      const size_t nb = sliceBase + (size_t)(band0 + b + 1) * BANDROWS * WDIM;
      const float4* xg = (const float4*)(x + nb);
      const float4* rg = (const float4*)(ref + nb);
#pragma unroll
      for (int c = 0; c < 8; ++c) { px[c] = xg[tid + c * 256]; pr[c] = rg[tid + c * 256]; }
    }

    // ---- row stats: Gram diagonals via fp32 WMMA ----
    v8f cXR = {}; v8f cXX = {}; v8f cRR = {};
#pragma unroll 4
    for (int kc = 0; kc < 16; ++kc) {
      const int c0 = cbase + (kc << 2) + koff;
      v2f a, bb;
      a.x  = xs[arow * LSTR + c0];
      a.y  = xs[arow * LSTR + c0 + 1];
      bb.x = rs[arow * LSTR + c0];
      bb.y = rs[arow * LSTR + c0 + 1];
      cXR = __builtin_amdgcn_wmma_f32_16x16x4_f32(false, a, false, bb, (short)0, cXR, false, false);
      cXX = __builtin_amdgcn_wmma_f32_16x16x4_f32(false, a, false, a,  (short)0, cXX, false, false);
      cRR = __builtin_amdgcn_wmma_f32_16x16x4_f32(false, bb, false, bb, (short)0, cRR, false, false);
    }

    // Diagonal of 16x16 f32 C: d[m] in (vgpr m, lane m) for m<8; (vgpr m-8, lane m+16) for m>=8.
    {
      int m = -1, r = 0;
      if (lane < 8)        { m = lane;      r = lane;      }
      else if (lane >= 24) { m = lane - 16; r = lane - 24; }
      if (m >= 0) {
        atomicAdd(&acc[m],      vext8(cXR, r));   // ds_add_f32
        atomicAdd(&acc[m + 16], vext8(cXX, r));
        atomicAdd(&acc[m + 32], vext8(cRR, r));
      }
    }

    // ---- column partials from the same LDS staging (into registers) ----
#pragma unroll
    for (int cc = 0; cc < 2; ++cc) {
      const int col = tid + cc * 256;
      float sd = 0.f, sx2 = 0.f, sr2 = 0.f;
#pragma unroll
      for (int rr2 = 0; rr2 < BANDROWS; ++rr2) {
        const float xv = xs[rr2 * LSTR + col];
        const float rv = rs[rr2 * LSTR + col];
        sd  = fmaf(xv, rv, sd);
        sx2 = fmaf(xv, xv, sx2);
        sr2 = fmaf(rv, rv, sr2);
      }
      csd[cc] += sd; csx[cc] += sx2; csr[cc] += sr2;
    }
    __syncthreads();

    // ---- per-band row finalize; owner threads read then re-zero (race-free:
    //      next band's ds atomics are ordered after the next barrier) ----
    if (tid < 16) {
      const float dot = acc[tid];
      const float xx  = acc[tid + 16];
      const float rr2 = acc[tid + 32];
      acc[tid] = 0.f; acc[tid + 16] = 0.f; acc[tid + 32] = 0.f;
      rowTermAcc += dot / (fmaxf(sqrtf(xx), EPSF) * fmaxf(sqrtf(rr2), EPSF));
    }
  }

  // ---- flush column partials ----
#pragma unroll
  for (int cc = 0; cc < 2; ++cc) {
    const int col = tid + cc * 256;
    if (ATOMIC_COLS) {
      const int base = slice * WDIM + col;
      gatomAdd(&colOut[base],                  csd[cc]);
      gatomAdd(&colOut[base + PLANE_SMALL],     csx[cc]);
      gatomAdd(&colOut[base + 2 * PLANE_SMALL], csr[cc]);
    } else {
      const int base = (slice * CHUNKS + chunk) * WDIM + col;   // uniquely owned
      colOut[base]                 = csd[cc];
      colOut[base + PLANE_BIG]     = csx[cc];
      colOut[base + 2 * PLANE_BIG] = csr[cc];
    }
  }

  // ---- block row-term reduce -> 1 atomic ----
  if (tid < 16) acc[tid] = rowTermAcc;   // same owner threads as above
  __syncthreads();
  if (tid == 0) {
    float s = 0.f;
#pragma unroll
    for (int i2 = 0; i2 < 16; ++i2) s += acc[i2];
    gatomAdd(rowSum, s);
  }
}

template <bool ATOMIC_COLS>
__global__ void __launch_bounds__(256) col_final_kernel(
    const float* __restrict__ colOut, float* __restrict__ colSum) {
  __shared__ float red[256];
  const int idx = blockIdx.x * 256 + threadIdx.x;   // (slice, col) pair
  float term = 0.f;
  if (idx < NSLICE * WDIM) {
    float sd, sx, sr;
    if (ATOMIC_COLS) {
      sd = colOut[idx];
      sx = colOut[idx + PLANE_SMALL];
      sr = colOut[idx + 2 * PLANE_SMALL];
    } else {
      const int slice = idx / WDIM;
      const int col   = idx % WDIM;
      sd = 0.f; sx = 0.f; sr = 0.f;
#pragma unroll
      for (int c = 0; c < CHUNKS; ++c) {
        const int base = (slice * CHUNKS + c) * WDIM + col;    // coalesced across threads
        sd += colOut[base];
        sx += colOut[base + PLANE_BIG];
        sr += colOut[base + 2 * PLANE_BIG];
      }
    }
    term = sd / (fmaxf(sqrtf(sx), EPSF) * fmaxf(sqrtf(sr), EPSF));
  }
  red[threadIdx.x] = term;
  __syncthreads();
#pragma unroll
  for (int s = 128; s > 0; s >>= 1) {
    if (threadIdx.x < s) red[threadIdx.x] += red[threadIdx.x + s];
    __syncthreads();
  }
  if (threadIdx.x == 0) gatomAdd(colSum, red[0]);
}

__global__ void final_kernel(const float* rowSum, const float* colSum, float* out) {
  // a-term: -sum(rowterm)/(H*B); b-term: -sum(colterm)/(W*B); H=W=512, B=16.
  out[0] = -(rowSum[0] + colSum[0]) * (1.0f / 8192.0f);
}

extern "C" void kernel_launch(void* const* d_in, const int* in_sizes, int n_in,
                              void* d_out, int out_size, void* d_ws, size_t ws_size,
                              hipStream_t stream) {
  const float* x   = (const float*)d_in[0];
  const float* ref = (const float*)d_in[1];
  float* out = (float*)d_out;
  float* ws  = (float*)d_ws;

  const size_t needBig = (size_t)(3 * PLANE_BIG + 2) * sizeof(float);   // ~4.72 MB
  const size_t shmem = (size_t)(2 * BANDROWS * LSTR + 48) * sizeof(float);
  const int nBlocks = NSLICE * CHUNKS;   // 768

  if (ws_size >= needBig) {
    // Store path: zero column atomics; partial planes fully overwritten each call.
    float* colOut = ws;
    float* rowSum = ws + 3 * PLANE_BIG;
    float* colSum = rowSum + 1;
    init_ws_kernel<<<1, 256, 0, stream>>>(rowSum, 2);
    band_kernel<false><<<nBlocks, 256, shmem, stream>>>(x, ref, colOut, rowSum);
    col_final_kernel<false><<<(NSLICE * WDIM) / 256, 256, 0, stream>>>(colOut, colSum);
    final_kernel<<<1, 1, 0, stream>>>(rowSum, colSum, out);
  } else {
    // Fallback: atomic accumulation into compact planes (288 KB scratch).
    float* colOut = ws;
    float* rowSum = ws + 3 * PLANE_SMALL;
    float* colSum = rowSum + 1;
    const int nzero = 3 * PLANE_SMALL + 2;
    init_ws_kernel<<<(nzero + 255) / 256, 256, 0, stream>>>(ws, nzero);
    band_kernel<true><<<nBlocks, 256, shmem, stream>>>(x, ref, colOut, rowSum);
    col_final_kernel<true><<<(NSLICE * WDIM) / 256, 256, 0, stream>>>(colOut, colSum);
    final_kernel<<<1, 1, 0, stream>>>(rowSum, colSum, out);
  }
}